// GroupedQueryAttention_2121713845118
// MI455X (gfx1250) — compile-verified
//
#include <hip/hip_runtime.h>
#include <hip/hip_bf16.h>

typedef __attribute__((ext_vector_type(2))) float v2f;
typedef __attribute__((ext_vector_type(8))) float v8f;
typedef __attribute__((ext_vector_type(4))) unsigned int v4u;
typedef __attribute__((ext_vector_type(8))) int v8i;
typedef __attribute__((ext_vector_type(4))) int v4i;

#define D_MODEL 2048
#define N_HEADS 16
#define N_KV 4
#define HEAD_DIM 128
#define SEQ 2048
#define BATCH 2
#define QKV_OUT ((N_HEADS + 2*N_KV)*HEAD_DIM)   // 3072
#define M_TOT (BATCH*SEQ)                        // 4096

#define HAVE_TDM (__has_builtin(__builtin_amdgcn_tensor_load_to_lds) && \
                  __has_builtin(__builtin_amdgcn_s_wait_tensorcnt))

__device__ __forceinline__ v8f wmma_f32(v2f a, v2f b, v8f c) {
  // V_WMMA_F32_16X16X4_F32 : D = A(16x4) * B(4x16) + C(16x16), all fp32
  return __builtin_amdgcn_wmma_f32_16x16x4_f32(false, a, false, b, (short)0, c,
                                               false, false);
}
__device__ __forceinline__ v8f v8zero() {
  v8f z = {0.f,0.f,0.f,0.f,0.f,0.f,0.f,0.f};
  return z;
}

#if HAVE_TDM
__device__ __forceinline__ unsigned lds_off(const void* p) {
  // generic LDS pointer: low 32 bits are the wave-relative LDS byte address
  return (unsigned)(unsigned long long)(uintptr_t)p;
}
// Issue one 2D TENSOR_LOAD_TO_LDS. data_size = 4B. Tile rows of tile_d0
// DWORDs are written to LDS with pad_amount DWORDs of padding appended every
// (1<<pad_int_code)*2 DWORDs, reproducing our bank-conflict row padding.
__device__ __forceinline__ void tdm_load_2d(unsigned lds_addr, const void* gaddr,
    unsigned tile_d0, unsigned tile_d1, unsigned tensor_d0, unsigned tensor_d1,
    unsigned long long stride_d0, unsigned pad_int_code, unsigned pad_amt_code)
{
  unsigned long long ga = (unsigned long long)(uintptr_t)gaddr;
  v4u g0;
  g0[0] = 1u;                                        // count=1 (valid user D#)
  g0[1] = lds_addr;                                  // lds_addr [63:32]
  g0[2] = (unsigned)(ga & 0xFFFFFFFFu);              // global_addr [95:64]
  g0[3] = (unsigned)((ga >> 32) & 0x01FFFFFFu)       // global_addr [120:96]
        | (2u << 30);                                // type=2 ("image") [127:126]
  v8i g1;
  g1[0] = (int)((2u << 16)                           // data_size = 4 bytes
              | (1u << 20)                           // pad_enable
              | (pad_int_code << 22)                 // pad_interval
              | (pad_amt_code << 25));               // pad_amount
  g1[1] = (int)((tensor_d0 & 0xFFFFu) << 16);        // tensor_dim0[15:0] @bit48
  g1[2] = (int)(((tensor_d0 >> 16) & 0xFFFFu)        // tensor_dim0[31:16]
              | ((tensor_d1 & 0xFFFFu) << 16));      // tensor_dim1[15:0] @bit80
  g1[3] = (int)(((tensor_d1 >> 16) & 0xFFFFu)        // tensor_dim1[31:16]
              | ((tile_d0 & 0xFFFFu) << 16));        // tile_dim0 @bit112
  g1[4] = (int)(tile_d1 & 0xFFFFu);                  // tile_dim1; tile_dim2=0
  g1[5] = (int)(unsigned)(stride_d0 & 0xFFFFFFFFull);// tensor_dim0_stride[31:0]
  g1[6] = (int)(unsigned)((stride_d0 >> 32) & 0xFFFFull);
  g1[7] = 0;
  v4i gz4 = {0, 0, 0, 0};
  v8i gz8 = {0, 0, 0, 0, 0, 0, 0, 0};
  // 6-arg toolchain form: (g0, g1, g2, g3, g_extra, cpol); 2D tensor ->
  // groups 2/3 (and the extra group) are zero.
  __builtin_amdgcn_tensor_load_to_lds(g0, g1, gz4, gz4, gz8, 0);
}
#endif

// ---------------------------------------------------------------------------
// Generic GEMM:  C[M,N] = A[M,K] * W[N,K]^T     (fp32, WMMA f32 16x16x4)
// Block tile 64(M) x 128(N), K-chunk 32, double buffered. 256 threads = 8
// waves (2x4), each wave owns a 32x32 sub-tile (four 16x16 WMMA accums).
// A tile staged by TDM (wave 0 issues descriptor, padded rows of 36 floats);
// B tile staged manually, pair-interleaved so each B frag is one ds_load_b64.
// ---------------------------------------------------------------------------
__global__ void __launch_bounds__(256)
gemm_nt_kernel(const float* __restrict__ A, const float* __restrict__ W,
               float* __restrict__ C, int M, int N, int K)
{
#if HAVE_TDM
  __shared__ float As[2][64][36];
  __shared__ v2f   BsP[2][16][130];   // [k/2][n] = {W[n][2k'], W[n][2k'+1]}
#else
  __shared__ float As[1][64][36];
  __shared__ v2f   BsP[1][16][130];
#endif

  const int tid  = threadIdx.x;
  const int lane = tid & 31;
  const int wv   = tid >> 5;
  const int wm   = wv >> 2;       // 0..1 (M)
  const int wn   = wv & 3;        // 0..3 (N)
  const int lm   = lane & 15;
  const int half = lane >> 4;

  const int Mbase = blockIdx.y * 64;
  const int Nbase = blockIdx.x * 128;

  v8f acc[2][2];
#pragma unroll
  for (int mi = 0; mi < 2; ++mi)
#pragma unroll
    for (int ni = 0; ni < 2; ++ni) acc[mi][ni] = v8zero();

  const int bn = tid >> 1;            // 0..127
  const int bk = (tid & 1) * 16;      // 0 or 16
  const int nchunks = K / 32;

  // manual B staging into pair-interleaved layout
  auto stage_B = [&](int buf, int k0) {
    const float4* wg = reinterpret_cast<const float4*>(
        &W[(size_t)(Nbase + bn) * K + k0 + bk]);
    float4 w0 = wg[0], w1 = wg[1], w2 = wg[2], w3 = wg[3];
    float* bp = (float*)&BsP[buf][0][0];
#define PUTB(kk, val) bp[(((bk + (kk)) >> 1) * 130 + bn) * 2 + ((bk + (kk)) & 1)] = (val)
    PUTB( 0, w0.x); PUTB( 1, w0.y); PUTB( 2, w0.z); PUTB( 3, w0.w);
    PUTB( 4, w1.x); PUTB( 5, w1.y); PUTB( 6, w1.z); PUTB( 7, w1.w);
    PUTB( 8, w2.x); PUTB( 9, w2.y); PUTB(10, w2.z); PUTB(11, w2.w);
    PUTB(12, w3.x); PUTB(13, w3.y); PUTB(14, w3.z); PUTB(15, w3.w);
#undef PUTB
  };

#if HAVE_TDM
  // prologue: prefetch chunk 0
  if (wv == 0)
    tdm_load_2d(lds_off(&As[0][0][0]), &A[(size_t)Mbase * K + 0],
                /*tile*/32, 64, /*tensor*/32, 64, (unsigned long long)K,
                /*pad: every 32 dwords*/4, /*+4 dwords*/3);
  stage_B(0, 0);
#else
  const int arow = tid >> 2;          // 0..63
  const int acol = (tid & 3) * 8;     // 0,8,16,24
#endif

  for (int ic = 0; ic < nchunks; ++ic) {
    const int k0 = ic * 32;
#if HAVE_TDM
    const int p  = ic & 1;
    const int nb = p ^ 1;
    if (ic + 1 < nchunks) {
      if (wv == 0)
        tdm_load_2d(lds_off(&As[nb][0][0]), &A[(size_t)Mbase * K + (k0 + 32)],
                    32, 64, 32, 64, (unsigned long long)K, 4, 3);
      stage_B(nb, k0 + 32);
      if (wv == 0) __builtin_amdgcn_s_wait_tensorcnt(1);  // chunk ic done
    } else {
      if (wv == 0) __builtin_amdgcn_s_wait_tensorcnt(0);
    }
    __syncthreads();
#else
    const int p = 0;
    {
      const float4* ag = reinterpret_cast<const float4*>(
          &A[(size_t)(Mbase + arow) * K + k0 + acol]);
      float4 a0 = ag[0], a1 = ag[1];
      *reinterpret_cast<float4*>(&As[0][arow][acol])     = a0;
      *reinterpret_cast<float4*>(&As[0][arow][acol + 4]) = a1;
      stage_B(0, k0);
    }
    __syncthreads();
#endif

#pragma unroll
    for (int ks = 0; ks < 8; ++ks) {
      const int kk = ks * 4 + 2 * half;     // lane's K pair base (even)
      // A frags: lane holds M=lm, K=kk+{0,1}  (contiguous -> ds_load_b64)
      v2f a0f = *reinterpret_cast<const v2f*>(&As[p][wm*32 +      lm][kk]);
      v2f a1f = *reinterpret_cast<const v2f*>(&As[p][wm*32 + 16 + lm][kk]);
      // B frags: lane holds N, K=kk+{0,1}   (pair-interleaved -> one b64)
      v2f b0f = BsP[p][ks*2 + half][wn*32 +      lm];
      v2f b1f = BsP[p][ks*2 + half][wn*32 + 16 + lm];
      acc[0][0] = wmma_f32(a0f, b0f, acc[0][0]);
      acc[0][1] = wmma_f32(a0f, b1f, acc[0][1]);
      acc[1][0] = wmma_f32(a1f, b0f, acc[1][0]);
      acc[1][1] = wmma_f32(a1f, b1f, acc[1][1]);
    }
    __syncthreads();
  }

#pragma unroll
  for (int mi = 0; mi < 2; ++mi)
#pragma unroll
    for (int ni = 0; ni < 2; ++ni)
#pragma unroll
      for (int v = 0; v < 8; ++v) {
        const int row = Mbase + wm*32 + mi*16 + v + 8*half;
        const int col = Nbase + wn*32 + ni*16 + lm;
        C[(size_t)row * N + col] = acc[mi][ni][v];
      }
}

// ---------------------------------------------------------------------------
// QKV post-processing: RMSNorm (q,k) + RoPE (q,k) + scatter to [b,h,s,d].
// One 128-thread block per (token m, 128-wide head unit hu).
// ---------------------------------------------------------------------------
__global__ void __launch_bounds__(128)
qkv_post_kernel(const float* __restrict__ qkv, const float* __restrict__ qnw,
                const float* __restrict__ knw, float* __restrict__ qb,
                float* __restrict__ kb, float* __restrict__ vb)
{
  __shared__ float red[128];
  __shared__ float row[128];
  const int m   = blockIdx.x;      // 0..4095
  const int hu  = blockIdx.y;      // 0..23
  const int t   = threadIdx.x;     // 0..127
  const int b   = m / SEQ;
  const int tok = m % SEQ;

  float val = qkv[(size_t)m * QKV_OUT + hu * HEAD_DIM + t];

  if (hu >= N_HEADS + N_KV) {      // V head: raw copy
    const int vh = hu - (N_HEADS + N_KV);
    vb[(((size_t)(b * N_KV + vh)) * SEQ + tok) * HEAD_DIM + t] = val;
    return;
  }

  red[t] = val * val;
  __syncthreads();
  for (int st = 64; st > 0; st >>= 1) {
    if (t < st) red[t] += red[t + st];
    __syncthreads();
  }
  const float rs = rsqrtf(red[0] * (1.0f / HEAD_DIM) + 1e-5f);
  const float* nw = (hu < N_HEADS) ? qnw : knw;
  row[t] = val * rs * nw[t];
  __syncthreads();

  const int   j    = t & 63;
  const float x1   = row[j];
  const float x2   = row[j + 64];
  const float invf = __expf(-(float)j * 0.14391156831212787f);  // ln(1e4)/64
  float sn, cs;
  sincosf((float)tok * invf, &sn, &cs);
  const float o = (t < 64) ? (x1 * cs - x2 * sn) : (x1 * sn + x2 * cs);

  if (hu < N_HEADS) {
    qb[(((size_t)(b * N_HEADS + hu)) * SEQ + tok) * HEAD_DIM + t] = o;
  } else {
    const int kh = hu - N_HEADS;
    kb[(((size_t)(b * N_KV + kh)) * SEQ + tok) * HEAD_DIM + t] = o;
  }
}

// ---------------------------------------------------------------------------
// Gate: sigmoid(x . w_gate[h] + b_gate[h]).
// ---------------------------------------------------------------------------
__global__ void __launch_bounds__(256)
gate_kernel(const float* __restrict__ x, const float* __restrict__ wg,
            const float* __restrict__ bg, float* __restrict__ gate)
{
  const int msk  = blockIdx.x;      // b*SEQ + s
  const int tid  = threadIdx.x;
  const int wv   = tid >> 5;
  const int lane = tid & 31;
  const int lm   = lane & 15;
  const int h    = wv * 2 + (lane >> 4);

  const float* xr = &x [(size_t)msk * D_MODEL];
  const float* wr = &wg[(size_t)h   * D_MODEL];
  float sum = 0.f;
  for (int i = lm; i < D_MODEL; i += 16) sum += xr[i] * wr[i];
#pragma unroll
  for (int off = 1; off < 16; off <<= 1) sum += __shfl_xor(sum, off, 32);
  if (lm == 0)
    gate[(size_t)msk * N_HEADS + h] = 1.f / (1.f + __expf(-(sum + bg[h])));
}

// ---------------------------------------------------------------------------
// Flash attention, causal, GQA (n_rep=4). Grid (S/128, H, B); 256 threads.
// Each wave owns 16 queries; Q A-frags in registers; K/V 16x128 tiles
// TDM-prefetched into double-buffered LDS (padded rows of 132 floats);
// online softmax; P re-laid out C->A via per-wave LDS scratch.
// ---------------------------------------------------------------------------
__global__ void __launch_bounds__(256)
attn_kernel(const float* __restrict__ qb, const float* __restrict__ kb,
            const float* __restrict__ vb, const float* __restrict__ gate,
            float* __restrict__ attn)
{
#if HAVE_TDM
  __shared__ float ks[2][16][132];
  __shared__ float vs[2][16][132];
#else
  __shared__ float ks[1][16][132];
  __shared__ float vs[1][16][132];
#endif
  __shared__ float Ps[8][16][18];

  const int tid  = threadIdx.x;
  const int wv   = tid >> 5;
  const int lane = tid & 31;
  const int lm   = lane & 15;
  const int half = lane >> 4;

  const int q0  = blockIdx.x * 128;
  const int h   = blockIdx.y;
  const int b   = blockIdx.z;
  const int kvh = h >> 2;           // n_rep = N_HEADS / N_KV = 4

  const float* qp = qb + ((size_t)(b * N_HEADS + h  )) * SEQ * HEAD_DIM;
  const float* kp = kb + ((size_t)(b * N_KV    + kvh)) * SEQ * HEAD_DIM;
  const float* vp = vb + ((size_t)(b * N_KV    + kvh)) * SEQ * HEAD_DIM;

  const int qw = q0 + wv * 16;

  v2f aq[32];
#pragma unroll
  for (int kk = 0; kk < 32; ++kk)
    aq[kk] = *reinterpret_cast<const v2f*>(
        &qp[(size_t)(qw + lm) * HEAD_DIM + kk * 4 + 2 * half]);

  float mrow[8], lrow[8], sf[8];
  v8f acc[8];
#pragma unroll
  for (int v = 0; v < 8; ++v) { mrow[v] = -3.0e38f; lrow[v] = 0.f; }
#pragma unroll
  for (int nT = 0; nT < 8; ++nT) acc[nT] = v8zero();

  const int numkb = q0 / 16 + 8;

#if HAVE_TDM
  if (wv == 0) {   // prefetch key-block 0 (K then V; same-wave TDM is in-order)
    tdm_load_2d(lds_off(&ks[0][0][0]), kp, 128, 16, 128, 16, 128ull,
                /*pad every 128 dwords*/6, /*+4 dwords*/3);
    tdm_load_2d(lds_off(&vs[0][0][0]), vp, 128, 16, 128, 16, 128ull, 6, 3);
  }
#else
  const int lrow_t = tid >> 4;      // 0..15
  const int lcol   = (tid & 15) * 8;
#endif

  for (int kb_i = 0; kb_i < numkb; ++kb_i) {
    const int kbase = kb_i * 16;
#if HAVE_TDM
    const int p  = kb_i & 1;
    const int nb = p ^ 1;
    if (kb_i + 1 < numkb) {
      if (wv == 0) {
        const size_t noff = (size_t)(kbase + 16) * HEAD_DIM;
        tdm_load_2d(lds_off(&ks[nb][0][0]), kp + noff, 128, 16, 128, 16, 128ull, 6, 3);
        tdm_load_2d(lds_off(&vs[nb][0][0]), vp + noff, 128, 16, 128, 16, 128ull, 6, 3);
        __builtin_amdgcn_s_wait_tensorcnt(2);   // block kb_i's pair complete
      }
    } else {
      if (wv == 0) __builtin_amdgcn_s_wait_tensorcnt(0);
    }
    __syncthreads();
#else
    const int p = 0;
    {
      const float4* kg = reinterpret_cast<const float4*>(
          &kp[(size_t)(kbase + lrow_t) * HEAD_DIM + lcol]);
      float4 k0v = kg[0], k1v = kg[1];
      *reinterpret_cast<float4*>(&ks[0][lrow_t][lcol])     = k0v;
      *reinterpret_cast<float4*>(&ks[0][lrow_t][lcol + 4]) = k1v;
      const float4* vg = reinterpret_cast<const float4*>(
          &vp[(size_t)(kbase + lrow_t) * HEAD_DIM + lcol]);
      float4 v0v = vg[0], v1v = vg[1];
      *reinterpret_cast<float4*>(&vs[0][lrow_t][lcol])     = v0v;
      *reinterpret_cast<float4*>(&vs[0][lrow_t][lcol + 4]) = v1v;
    }
    __syncthreads();
#endif

    // scores = q (16x128) . k^T (128x16)
    v8f sc = v8zero();
#pragma unroll
    for (int kk = 0; kk < 32; ++kk) {
      v2f bk = *reinterpret_cast<const v2f*>(&ks[p][lm][kk * 4 + 2 * half]);
      sc = wmma_f32(aq[kk], bk, sc);
    }

    // online softmax (C layout: row = v + 8*half, col/key = lm)
#pragma unroll
    for (int v = 0; v < 8; ++v) {
      const int qrow  = qw + v + 8 * half;
      const int key   = kbase + lm;
      const float s   = sc[v] * 0.08838834764831845f;  // 1/sqrt(128)
      const bool  ok  = key <= qrow;
      const float smv = ok ? s : -3.0e38f;
      float rm = smv;
#pragma unroll
      for (int off = 1; off < 16; off <<= 1) rm = fmaxf(rm, __shfl_xor(rm, off, 32));
      const float nm = fmaxf(mrow[v], rm);
      const float pv = ok ? __expf(s - nm) : 0.f;
      float rsum = pv;
#pragma unroll
      for (int off = 1; off < 16; off <<= 1) rsum += __shfl_xor(rsum, off, 32);
      sf[v]   = __expf(mrow[v] - nm);
      lrow[v] = lrow[v] * sf[v] + rsum;
      mrow[v] = nm;
      Ps[wv][v + 8 * half][lm] = pv;   // C-layout -> LDS -> A-layout reload
    }
#pragma unroll
    for (int nT = 0; nT < 8; ++nT)
#pragma unroll
      for (int v = 0; v < 8; ++v) acc[nT][v] *= sf[v];

    // acc += P (16x16) . V (16x128)
#pragma unroll
    for (int k4 = 0; k4 < 4; ++k4) {
      const int kr = k4 * 4 + 2 * half;
      v2f ap = *reinterpret_cast<const v2f*>(&Ps[wv][lm][kr]);
#pragma unroll
      for (int nT = 0; nT < 8; ++nT) {
        v2f bvv;
        bvv.x = vs[p][kr    ][nT * 16 + lm];
        bvv.y = vs[p][kr + 1][nT * 16 + lm];
        acc[nT] = wmma_f32(ap, bvv, acc[nT]);
      }
    }
    __syncthreads();   // done reading buf p; TDM may overwrite it next iter
  }

  // epilogue: 1/l, gate, store gated attention [b,s, h*128+d]
#pragma unroll
  for (int v = 0; v < 8; ++v) {
    const int   qrow  = qw + v + 8 * half;
    const float inv   = (lrow[v] > 0.f) ? 1.f / lrow[v] : 0.f;
    const float g     = gate[(size_t)(b * SEQ + qrow) * N_HEADS + h];
    const float scale = inv * g;
    float* orow = &attn[((size_t)(b * SEQ + qrow)) * D_MODEL + h * HEAD_DIM];
#pragma unroll
    for (int nT = 0; nT < 8; ++nT) orow[nT * 16 + lm] = acc[nT][v] * scale;
  }
}

// ---------------------------------------------------------------------------
extern "C" void kernel_launch(void* const* d_in, const int* in_sizes, int n_in,
                              void* d_out, int out_size, void* d_ws, size_t ws_size,
                              hipStream_t stream)
{
  (void)in_sizes; (void)n_in; (void)out_size; (void)ws_size;

  const float* x    = (const float*)d_in[0];
  const float* wqkv = (const float*)d_in[1];
  const float* qnw  = (const float*)d_in[2];
  const float* knw  = (const float*)d_in[3];
  const float* wg   = (const float*)d_in[4];
  const float* bg   = (const float*)d_in[5];
  const float* wo   = (const float*)d_in[6];
  float* out = (float*)d_out;
  float* ws  = (float*)d_ws;

  float* qkv_ws  = ws;                                    // 4096*3072 f32
  float* attn_ws = ws;                                    // reuses qkv region
  float* q_ws = ws   + (size_t)M_TOT * QKV_OUT;
  float* k_ws = q_ws + (size_t)BATCH * N_HEADS * SEQ * HEAD_DIM;
  float* v_ws = k_ws + (size_t)BATCH * N_KV   * SEQ * HEAD_DIM;
  float* g_ws = v_ws + (size_t)BATCH * N_KV   * SEQ * HEAD_DIM;

  gemm_nt_kernel<<<dim3(QKV_OUT / 128, M_TOT / 64), 256, 0, stream>>>(
      x, wqkv, qkv_ws, M_TOT, QKV_OUT, D_MODEL);
  qkv_post_kernel<<<dim3(M_TOT, N_HEADS + 2 * N_KV), 128, 0, stream>>>(
      qkv_ws, qnw, knw, q_ws, k_ws, v_ws);
  gate_kernel<<<dim3(M_TOT), 256, 0, stream>>>(x, wg, bg, g_ws);
  attn_kernel<<<dim3(SEQ / 128, N_HEADS, BATCH), 256, 0, stream>>>(
      q_ws, k_ws, v_ws, g_ws, attn_ws);
  gemm_nt_kernel<<<dim3(D_MODEL / 128, M_TOT / 64), 256, 0, stream>>>(
      attn_ws, wo, out, M_TOT, D_MODEL, D_MODEL);
}